// DynamicDilationUnfold_53764400611512
// MI455X (gfx1250) — compile-verified
//
#include <hip/hip_runtime.h>
#include <hip/hip_bf16.h>

// DynamicDilationUnfold on MI455X (gfx1250, wave32).
//
// Shapes (compile-time constants from the reference):
//   x    : (B=8, C=64, H=128, W=128) fp32   (33.6 MB, fully L2-resident)
//   dmap : (B, 1, 128, 128) fp32
//   out  : (B, C*9, Ho*Wo) = (8, 576, 16384) fp32  (~302 MB, write-once)
//
// Roofline: ~0.6 GFLOP vs ~302 MB streaming stores + cached gathers ->
// store-bandwidth bound (~15-25 us at 23.3 TB/s). No matmul structure
// (corner values vary per output row), so WMMA does not apply. Strategy:
//   * amortize bilinear weight math over the 64-channel loop,
//   * coalesced lane->wo stores with NT temporal hint (keep L2 for x),
//   * unroll-8 channel loop for deep load pipelining into L2/WGP$.

#define BN   8
#define CN   64
#define HN   128
#define WN   128
#define HO   128
#define WO   128
#define HW   (HN * WN)     // 16384 floats per channel plane
#define PIX  (HO * WO)     // 16384 output pixels per (b, channel-row)

__global__ __launch_bounds__(256, 8)
void DynamicDilationUnfold_53764400611512_kernel(const float* __restrict__ x,
                                                 const float* __restrict__ dmap,
                                                 float* __restrict__ out)
{
    // Grid layout: blockIdx.x = ((b*9 + kidx) * 64 + pixel_block)
    //   pixel_block covers 256 consecutive output pixels (2 rows of 128).
    // Lanes are consecutive in wo -> coalesced stores (128B per wave32).
    const int pix  = (blockIdx.x & 63) * 256 + threadIdx.x;  // ho*WO + wo
    const int wo   = pix & (WO - 1);
    const int ho   = pix >> 7;
    const int bk   = blockIdx.x >> 6;   // b*9 + kidx  (scalar, uniform)
    const int kidx = bk % 9;
    const int b    = bk / 9;

    const int ki = kidx / 3;
    const int kj = kidx % 3;

    // Per-pixel dilation (DIL=1, STRIDE=1, PAD=1 folded in as constants).
    const float d = dmap[b * PIX + pix];

    const float pos_r = (float)(ho - 1) + (float)ki * d;
    const float pos_c = (float)(wo - 1) + (float)kj * d;

    const float r0f = floorf(pos_r);
    const float c0f = floorf(pos_c);
    const float wr  = pos_r - r0f;
    const float wc  = pos_c - c0f;

    const int r0 = (int)r0f;
    const int c0 = (int)c0f;
    const int r1 = r0 + 1;
    const int c1 = c0 + 1;

    // Fold the out-of-bounds masks directly into the bilinear weights:
    // corner(img,r,c) = img[clip(r),clip(c)] * valid(r,c) and
    // valid(r,c) == valid_r(r) * valid_c(c) for 0/1 masks.
    const float vr0 = (r0 >= 0 && r0 < HN) ? 1.0f : 0.0f;
    const float vr1 = (r1 >= 0 && r1 < HN) ? 1.0f : 0.0f;
    const float vc0 = (c0 >= 0 && c0 < WN) ? 1.0f : 0.0f;
    const float vc1 = (c1 >= 0 && c1 < WN) ? 1.0f : 0.0f;

    const float w00 = (1.0f - wr) * (1.0f - wc) * vr0 * vc0;
    const float w01 = (1.0f - wr) * wc          * vr0 * vc1;
    const float w10 = wr          * (1.0f - wc) * vr1 * vc0;
    const float w11 = wr          * wc          * vr1 * vc1;

    const int r0c = min(max(r0, 0), HN - 1);
    const int r1c = min(max(r1, 0), HN - 1);
    const int c0c = min(max(c0, 0), WN - 1);
    const int c1c = min(max(c1, 0), WN - 1);

    // Four clamped corner base pointers for channel 0; per-channel stride is
    // the compile-time constant HW -> inner loop is pure add + load.
    const float* __restrict__ xb  = x + (size_t)b * CN * HW;
    const float* __restrict__ p00 = xb + r0c * WN + c0c;
    const float* __restrict__ p01 = xb + r0c * WN + c1c;
    const float* __restrict__ p10 = xb + r1c * WN + c0c;
    const float* __restrict__ p11 = xb + r1c * WN + c1c;

    // out[b][c*9 + kidx][pix]; stride between channels is 9*PIX.
    float* __restrict__ po = out + ((size_t)b * (CN * 9) + kidx) * PIX + pix;

    // Prime the gather stream (gfx1250 global_prefetch_b8).
    __builtin_prefetch(p00, 0, 1);
    __builtin_prefetch(p10, 0, 1);

    // 64-channel loop: 4 cached loads + 4 FMA + 1 coalesced NT store per
    // iteration. NT store (CDNA5 TH_STORE_NT) streams the write-once output
    // past the caches so L2/WGP$ stay dedicated to x (33.6 MB, ~36x reuse).
    #pragma unroll 8
    for (int c = 0; c < CN; ++c) {
        const size_t ci = (size_t)c * HW;
        const float a00 = p00[ci];
        const float a01 = p01[ci];
        const float a10 = p10[ci];
        const float a11 = p11[ci];
        const float r = a00 * w00 + a01 * w01 + a10 * w10 + a11 * w11;
        __builtin_nontemporal_store(r, po + (size_t)c * (9 * PIX));
    }
}

extern "C" void kernel_launch(void* const* d_in, const int* in_sizes, int n_in,
                              void* d_out, int out_size, void* d_ws, size_t ws_size,
                              hipStream_t stream) {
    (void)in_sizes; (void)n_in; (void)d_ws; (void)ws_size; (void)out_size;

    const float* x    = (const float*)d_in[0];   // (8, 64, 128, 128) fp32
    const float* dmap = (const float*)d_in[1];   // (8, 1, 128, 128) fp32
    float*       out  = (float*)d_out;           // (8, 576, 16384) fp32

    // One thread per (b, kidx, ho, wo); each loops over the 64 channels.
    // Blocks: (B * 9 taps) * (PIX / 256) = 72 * 64 = 4608 blocks of 256.
    const int blocks = BN * 9 * (PIX / 256);
    DynamicDilationUnfold_53764400611512_kernel<<<blocks, 256, 0, stream>>>(x, dmap, out);
}